// CTCInferenceLayer_55516747268899
// MI455X (gfx1250) — compile-verified
//
#include <hip/hip_runtime.h>
#include <hip/hip_bf16.h>
#include <stdint.h>

#define T_STEPS 1024
#define BATCH   32
#define C_CLS   96
#define W_BEAM  16
#define P_TOP   4
#define NEGF    (-1.0e30f)

typedef __attribute__((ext_vector_type(2))) float v2f;
typedef __attribute__((ext_vector_type(8))) float v8f;

__device__ __forceinline__ float lse(float a, float b) {
    float mx = fmaxf(a, b);
    float mn = fminf(a, b);
    return mx + __logf(1.0f + __expf(mn - mx));
}

__global__ __launch_bounds__(32)
void ctc_beam_kernel(const float* __restrict__ data,      // [T, B, C] log-probs
                     const int*   __restrict__ dlen,      // [B]
                     float*       __restrict__ out) {     // [B*P | B*P | B*P*T]
    const int bb   = blockIdx.x;      // batch element
    const int lane = threadIdx.x;     // wave32
    const int seqlen = dlen[bb];

    __shared__ __align__(16) float          s_lp[2][C_CLS];       // double-buffered lp slice
    __shared__ unsigned short               s_bp[T_STEPS * W_BEAM]; // backptr|sym per (t,beam)

    // Beam state lives on lanes 0..15 (lane == beam index).
    float pb   = (lane == 0) ? 0.0f : NEGF;
    float pnb  = NEGF;
    int   last = -1;
    int   lens = 0;

    const float* gbase = data + (size_t)bb * C_CLS;

    // ---- prologue: async prefetch lp[t=0] into s_lp[0] (24 lanes x 16B = 96 floats)
    {
        unsigned long long gp = (unsigned long long)(uintptr_t)gbase; // t=0 slice base
        uint32_t ldsa = (uint32_t)(uintptr_t)(&s_lp[0][0]) + (uint32_t)lane * 16u;
        uint32_t voff = (uint32_t)lane * 16u;
        if (lane < 24) {
            asm volatile("global_load_async_to_lds_b128 %0, %1, %2"
                         :: "v"(ldsa), "v"(voff), "s"(gp) : "memory");
        }
    }

    for (int t = 0; t < T_STEPS; ++t) {
        const int buf = t & 1;
        // wait for this step's lp slice; ASYNCcnt completes in-order
        asm volatile("s_wait_asynccnt 0x0" ::: "memory");
        // kick off prefetch of next slice into the other buffer (overlaps compute)
        if (t + 1 < T_STEPS) {
            unsigned long long gp =
                (unsigned long long)(uintptr_t)(gbase + (size_t)(t + 1) * (BATCH * C_CLS));
            uint32_t ldsa = (uint32_t)(uintptr_t)(&s_lp[buf ^ 1][0]) + (uint32_t)lane * 16u;
            uint32_t voff = (uint32_t)lane * 16u;
            if (lane < 24) {
                asm volatile("global_load_async_to_lds_b128 %0, %1, %2"
                             :: "v"(ldsa), "v"(voff), "s"(gp) : "memory");
            }
        }
        const float* lpc = s_lp[buf];

        // ---- per-beam (lanes 0..15) stay scores
        float ptot     = lse(pb, pnb);
        float lp_blank = lpc[0];
        float lp_last  = lpc[last < 0 ? 0 : last];
        float stay_pb  = ptot + lp_blank;
        float stay_pnb = (lens > 0) ? (pnb + lp_last) : NEGF;
        float stay_tot = lse(stay_pb, stay_pnb);

        // ---- broadcast beam data needed for per-slot fixups
        // slot (tile, v) on this lane corresponds to beam = v + 8*(lane>=16)
        const int hi = (lane >= 16) ? 8 : 0;
        float pb_h[8];
        int   last_h[8];
        #pragma unroll
        for (int v = 0; v < 8; ++v) {
            pb_h[v]   = __shfl(pb,   v + hi);
            last_h[v] = __shfl(last, v + hi);
        }

        // ---- ext scores via WMMA outer-sum: D[b][c] = ptot[b]*1 + 1*lp[c]
        // A (16x4 f32): lanes0-15 {K0=ptot, K1=1}; lanes16-31 {K2=0, K3=0}
        v2f A;
        A.x = (lane < 16) ? ptot : 0.0f;
        A.y = (lane < 16) ? 1.0f : 0.0f;

        float cand[49];
        const int lm = lane & 15;   // class-within-tile for this lane (D N-index)
        #pragma unroll
        for (int tile = 0; tile < 6; ++tile) {
            float lpl = lpc[tile * 16 + lm];
            // B (4x16 f32): row0 = ones, row1 = lp tile, rows 2-3 = 0
            v2f Bm;
            Bm.x = (lane < 16) ? 1.0f : 0.0f;
            Bm.y = (lane < 16) ? lpl  : 0.0f;
            v8f Cz = {0.f, 0.f, 0.f, 0.f, 0.f, 0.f, 0.f, 0.f};
            v8f D = __builtin_amdgcn_wmma_f32_16x16x4_f32(
                        false, A, false, Bm, (short)0, Cz, false, false);
            #pragma unroll
            for (int v = 0; v < 8; ++v) {
                // D vgpr v on this lane = ext[beam = v+hi][class = tile*16+lm]
                float e = D[v];
                bool eq = ((tile * 16 + lm) == last_h[v]);   // c == last[b] -> use pb
                e = eq ? (pb_h[v] + lpl) : e;
                if (tile == 0) e = (lm == 0) ? NEGF : e;     // blank never extends
                cand[tile * 8 + v] = e;
            }
        }
        cand[48] = (lane < 16) ? stay_tot : NEGF;            // stay candidates

        // ---- top-16 selection (16 rounds of wave argmax)
        float    npb = NEGF, npnb = NEGF;
        int      nlast = -1, nlens = 0;
        unsigned nbp = 0;
        #pragma unroll 1
        for (int r = 0; r < W_BEAM; ++r) {
            // lane-local argmax over 49 slots
            float m = cand[0];
            int  mi = 0;
            #pragma unroll
            for (int j = 1; j < 49; ++j) {
                bool g = cand[j] > m;
                m  = g ? cand[j] : m;
                mi = g ? j : mi;
            }
            int id = (lane << 6) | mi;
            // butterfly max-reduction across the wave
            #pragma unroll
            for (int k = 16; k >= 1; k >>= 1) {
                float om  = __shfl_xor(m, k);
                int   oid = __shfl_xor(id, k);
                bool g = (om > m) || ((om == m) && (oid < id));
                m  = g ? om  : m;
                id = g ? oid : id;
            }
            // scalarize winner: all bookkeeping on SALU
            int sid = __builtin_amdgcn_readfirstlane(id);
            int swl = sid >> 6;
            int sws = sid & 63;
            bool is_stay = (sws == 48);
            int v    = sws & 7;
            int tile = sws >> 3;
            int beam = v + ((swl >= 16) ? 8 : 0);
            int cls  = tile * 16 + (swl & 15);
            int parent = is_stay ? swl : beam;
            // parent state (uniform-index shuffles from beam lanes)
            float s_pbv  = __shfl(stay_pb,  parent);
            float s_pnbv = __shfl(stay_pnb, parent);
            int   p_last = __shfl(last, parent);
            int   p_lens = __shfl(lens, parent);
            float r_pb   = is_stay ? s_pbv  : NEGF;
            float r_pnb  = is_stay ? s_pnbv : m;        // ext: pnb = winning ext score
            int   r_last = is_stay ? p_last : cls;
            int   r_lens = is_stay ? p_lens : (p_lens + 1);
            unsigned r_bp = (unsigned)parent | ((unsigned)(is_stay ? 255 : cls) << 8);
            if (lane == r) { npb = r_pb; npnb = r_pnb; nlast = r_last; nlens = r_lens; nbp = r_bp; }
            // kill winner (scalar-indexed, single divergent lane)
            if (lane == swl) {
                #pragma unroll
                for (int j = 0; j < 49; ++j)
                    if (j == sws) cand[j] = NEGF;
            }
        }

        // ---- commit (only while t < length), log backpointers
        if (lane < W_BEAM) s_bp[t * W_BEAM + lane] = (unsigned short)nbp;
        if (t < seqlen) { pb = npb; pnb = npnb; last = nlast; lens = nlens; }
    }

    // ---- epilogue: top-4 beams
    float tot = (lane < W_BEAM) ? lse(pb, pnb) : NEGF;
    float myscore = 0.0f;
    int   mybeam  = 0;
    #pragma unroll
    for (int p = 0; p < P_TOP; ++p) {
        float m = tot;
        int  id = lane;
        #pragma unroll
        for (int k = 16; k >= 1; k >>= 1) {
            float om  = __shfl_xor(m, k);
            int   oid = __shfl_xor(id, k);
            bool g = (om > m) || ((om == m) && (oid < id));
            m  = g ? om  : m;
            id = g ? oid : id;
        }
        int swl = __builtin_amdgcn_readfirstlane(id);
        if (lane == p) { myscore = m; mybeam = swl; }
        if (lane == swl) tot = NEGF;
    }
    int mylen = __shfl(lens, mybeam);   // executed by all lanes (shfl needs full EXEC)

    float* out_scores = out;                          // [B][P]
    float* out_lens   = out + BATCH * P_TOP;          // [B][P]
    float* out_lab    = out + 2 * BATCH * P_TOP;      // [B][P][T]
    if (lane < P_TOP) {
        out_scores[bb * P_TOP + lane] = -myscore;
        out_lens[bb * P_TOP + lane]   = (float)mylen;
    }

    // ---- reconstruct labels by walking backpointers (lanes 0..3 = paths)
    int cur = mybeam;
    int pos = mylen - 1;
    const size_t lab_base = ((size_t)bb * P_TOP + (size_t)lane) * T_STEPS;
    for (int t = seqlen - 1; t >= 0; --t) {
        unsigned e = s_bp[t * W_BEAM + cur];
        int sym = (int)(e >> 8);
        int par = (int)(e & 255u);
        if (lane < P_TOP && sym != 255 && pos >= 0) {
            out_lab[lab_base + pos] = (float)sym;
            pos--;
        }
        cur = par;
    }
}

__global__ void init_labels(float* __restrict__ out) {
    int i = blockIdx.x * blockDim.x + threadIdx.x;
    const int n = BATCH * P_TOP * T_STEPS;
    if (i < n) out[2 * BATCH * P_TOP + i] = -1.0f;   // reference default label = -1
}

extern "C" void kernel_launch(void* const* d_in, const int* in_sizes, int n_in,
                              void* d_out, int out_size, void* d_ws, size_t ws_size,
                              hipStream_t stream) {
    const float* data = (const float*)d_in[0];   // [T, B, C] float32 log-probs
    const int*   dlen = (const int*)d_in[1];     // [B] int32
    float* out = (float*)d_out;

    const int n_lab = BATCH * P_TOP * T_STEPS;
    init_labels<<<(n_lab + 255) / 256, 256, 0, stream>>>(out);
    ctc_beam_kernel<<<BATCH, 32, 0, stream>>>(data, dlen, out);
}